// Model_16638703304940
// MI455X (gfx1250) — compile-verified
//
#include <hip/hip_runtime.h>
#include <hip/hip_bf16.h>

#define N_NODES    100000
#define N_EDGES    640000
#define IN_DIM     64
#define HID        128
#define NUM_GRAPHS 128
#define MPNN_STEPS 3
#define S2S_STEPS  3

typedef __attribute__((ext_vector_type(2))) float v2f;
typedef __attribute__((ext_vector_type(8))) float v8f;

// CDNA5 async global->LDS copies (ASYNCcnt-tracked), guarded so the build
// stays green if this toolchain lacks the builtins.
#if defined(__AMDGCN__) && __has_builtin(__builtin_amdgcn_global_load_async_to_lds_b128) && __has_builtin(__builtin_amdgcn_s_wait_asynccnt)
#define USE_ASYNC_LDS 1

// Builtin signature (from hipcc diagnostic): param0 = global (AS1) pointer to
// a 16B int vector, param1 = LDS (AS3) pointer, then imm offset + imm cpol.
typedef int b128_t __attribute__((vector_size(16)));
typedef __attribute__((address_space(1))) b128_t* gb128p;
typedef __attribute__((address_space(3))) b128_t* lb128p;

__device__ __forceinline__ void async_copy16(const float* g, float* l) {
    __builtin_amdgcn_global_load_async_to_lds_b128(
        (gb128p)(void*)(g), (lb128p)(l), 0, 0);
}
#endif

__device__ __forceinline__ float sigmoidf_(float x) {
    return 1.0f / (1.0f + expf(-x));
}

// float atomic max via int punning (valid when slot initialized to -inf)
__device__ __forceinline__ void atomicMaxF(float* addr, float val) {
    if (val >= 0.0f) atomicMax((int*)addr, __float_as_int(val));
    else             atomicMin((unsigned int*)addr, __float_as_uint(val));
}

__global__ void fill_kernel(float* __restrict__ p, float v, int n) {
    int i = blockIdx.x * blockDim.x + threadIdx.x;
    if (i < n) p[i] = v;
}

// ---------------------------------------------------------------------------
// Input layer: h = relu(x @ W_in^T + b_in).  16 nodes/block, 8 waves, each
// wave owns a 16-wide output-feature tile; K=64 via f32 WMMA 16x16x4.
// ---------------------------------------------------------------------------
__global__ __launch_bounds__(256) void input_kernel(
    const float* __restrict__ x, const float* __restrict__ W_in,
    const float* __restrict__ b_in, float* __restrict__ h)
{
    __shared__ float s_x[16 * IN_DIM];
    const int node0 = blockIdx.x * 16;
    const int tid = threadIdx.x;

#if defined(USE_ASYNC_LDS)
    {   // 16*64 floats = 4KB = 256 lanes x 16B: one async b128 per thread
        async_copy16(x + (size_t)node0 * IN_DIM + tid * 4, s_x + tid * 4);
        __builtin_amdgcn_s_wait_asynccnt(0);
    }
#else
    ((float4*)s_x)[tid] = ((const float4*)(x + (size_t)node0 * IN_DIM))[tid];
#endif
    __syncthreads();

    const int wave = tid >> 5;
    const int lane = tid & 31;
    const int row  = lane & 15;
    const int kg   = (lane >> 4) << 1;           // 0 or 2
    const int col  = wave * 16 + (lane & 15);

    v8f acc = {};
#pragma unroll 4
    for (int kb = 0; kb < IN_DIM; kb += 4) {
        v2f a, b;
        a.x = s_x[row * IN_DIM + kb + kg];
        a.y = s_x[row * IN_DIM + kb + kg + 1];
        b.x = W_in[col * IN_DIM + kb + kg];
        b.y = W_in[col * IN_DIM + kb + kg + 1];
        acc = __builtin_amdgcn_wmma_f32_16x16x4_f32(false, a, false, b,
                                                    (short)0, acc, false, false);
    }
    const float bias = b_in[col];
#pragma unroll
    for (int r = 0; r < 8; ++r) {
        const int mr = r + ((lane >> 4) << 3);   // D layout: row r / r+8
        float v = acc[r] + bias;
        h[(size_t)(node0 + mr) * HID + col] = v > 0.0f ? v : 0.0f;
    }
}

// ---------------------------------------------------------------------------
// Edge aggregation: m[dst] += h[src]. One wave32 per edge, float4 per lane
// (512B coalesced gather per edge; scatter atomics land in the 192MB L2,
// since m is only 51.2MB).
// ---------------------------------------------------------------------------
__global__ void edge_agg_kernel(
    const float* __restrict__ h, const int* __restrict__ src,
    const int* __restrict__ dst, float* __restrict__ m, int n_edges)
{
    const int idx  = blockIdx.x * blockDim.x + threadIdx.x;
    const int e    = idx >> 5;
    const int lane = idx & 31;
    if (e >= n_edges) return;
    const int s = src[e], d = dst[e];
    const float4 v = ((const float4*)(h + (size_t)s * HID))[lane];
    float* o = m + (size_t)d * HID + lane * 4;
    atomicAdd(o + 0, v.x);
    atomicAdd(o + 1, v.y);
    atomicAdd(o + 2, v.z);
    atomicAdd(o + 3, v.w);
}

// ---------------------------------------------------------------------------
// Fused MGU step (updates h in place; MGU is row-local given m):
//   f  = sigmoid(m @ Wf^T + h @ Uf^T + bf)
//   g  = f * h                      (exchanged block-wide through LDS)
//   hh = tanh(m @ Wh^T + g @ Uh^T + bh)
//   h  = (1-f)*h + f*hh
// 16 nodes/block, 8 waves x 16-col WMMA tiles, K=128 via f32 WMMA 16x16x4.
// Node tiles staged into LDS with CDNA5 async-to-LDS copies when available.
// ---------------------------------------------------------------------------
__global__ __launch_bounds__(256) void mgu_kernel(
    const float* __restrict__ m, float* __restrict__ h,
    const float* __restrict__ Wf, const float* __restrict__ Uf,
    const float* __restrict__ bf,
    const float* __restrict__ Wh, const float* __restrict__ Uh,
    const float* __restrict__ bh)
{
    __shared__ float s_m[16 * HID];
    __shared__ float s_h[16 * HID];
    __shared__ float s_g[16 * HID];

    const int node0 = blockIdx.x * 16;
    const int tid = threadIdx.x;

#if defined(USE_ASYNC_LDS)
    {   // 2 x 8KB tiles; 256 lanes x 16B x 2 async b128 copies per tile.
        const float* gm = m + (size_t)node0 * HID;
        const float* gh = h + (size_t)node0 * HID;
        async_copy16(gm + tid * 4,         s_m + tid * 4);
        async_copy16(gm + (tid + 256) * 4, s_m + (tid + 256) * 4);
        async_copy16(gh + tid * 4,         s_h + tid * 4);
        async_copy16(gh + (tid + 256) * 4, s_h + (tid + 256) * 4);
        __builtin_amdgcn_s_wait_asynccnt(0);  // my LDS writes landed
    }
#else
    {   // cooperative load of node tiles: 2048 floats each -> 2 float4/thread
        const float4* gm = (const float4*)(m + (size_t)node0 * HID);
        const float4* gh = (const float4*)(h + (size_t)node0 * HID);
        float4* sm4 = (float4*)s_m;
        float4* sh4 = (float4*)s_h;
        sm4[tid]       = gm[tid];
        sm4[tid + 256] = gm[tid + 256];
        sh4[tid]       = gh[tid];
        sh4[tid + 256] = gh[tid + 256];
    }
#endif
    __syncthreads();

    const int wave = tid >> 5;
    const int lane = tid & 31;
    const int row  = lane & 15;
    const int kg   = (lane >> 4) << 1;
    const int col  = wave * 16 + (lane & 15);
    const int mrhi = (lane >> 4) << 3;

    // ---- GEMM pair 1: forget gate ----
    v8f acc = {};
#pragma unroll 4
    for (int kb = 0; kb < HID; kb += 4) {
        v2f a, b;
        a.x = s_m[row * HID + kb + kg];
        a.y = s_m[row * HID + kb + kg + 1];
        b.x = Wf[col * HID + kb + kg];
        b.y = Wf[col * HID + kb + kg + 1];
        acc = __builtin_amdgcn_wmma_f32_16x16x4_f32(false, a, false, b,
                                                    (short)0, acc, false, false);
    }
#pragma unroll 4
    for (int kb = 0; kb < HID; kb += 4) {
        v2f a, b;
        a.x = s_h[row * HID + kb + kg];
        a.y = s_h[row * HID + kb + kg + 1];
        b.x = Uf[col * HID + kb + kg];
        b.y = Uf[col * HID + kb + kg + 1];
        acc = __builtin_amdgcn_wmma_f32_16x16x4_f32(false, a, false, b,
                                                    (short)0, acc, false, false);
    }
    const float biasf = bf[col];
    float fgate[8];
#pragma unroll
    for (int r = 0; r < 8; ++r) {
        fgate[r] = sigmoidf_(acc[r] + biasf);
        const int mr = r + mrhi;
        s_g[mr * HID + col] = fgate[r] * s_h[mr * HID + col];
    }
    __syncthreads();

    // ---- GEMM pair 2: candidate ----
    v8f acc2 = {};
#pragma unroll 4
    for (int kb = 0; kb < HID; kb += 4) {
        v2f a, b;
        a.x = s_m[row * HID + kb + kg];
        a.y = s_m[row * HID + kb + kg + 1];
        b.x = Wh[col * HID + kb + kg];
        b.y = Wh[col * HID + kb + kg + 1];
        acc2 = __builtin_amdgcn_wmma_f32_16x16x4_f32(false, a, false, b,
                                                     (short)0, acc2, false, false);
    }
#pragma unroll 4
    for (int kb = 0; kb < HID; kb += 4) {
        v2f a, b;
        a.x = s_g[row * HID + kb + kg];
        a.y = s_g[row * HID + kb + kg + 1];
        b.x = Uh[col * HID + kb + kg];
        b.y = Uh[col * HID + kb + kg + 1];
        acc2 = __builtin_amdgcn_wmma_f32_16x16x4_f32(false, a, false, b,
                                                     (short)0, acc2, false, false);
    }
    const float biash = bh[col];
#pragma unroll
    for (int r = 0; r < 8; ++r) {
        const int mr = r + mrhi;
        const float hh = tanhf(acc2[r] + biash);
        const float hv = s_h[mr * HID + col];
        h[(size_t)(node0 + mr) * HID + col] =
            (1.0f - fgate[r]) * hv + fgate[r] * hh;
    }
}

// ---------------------------------------------------------------------------
// Set2Set LSTM cell: one block (512 threads) per graph; gate j per thread,
// then threads j<HID apply the cell update.
// ---------------------------------------------------------------------------
__global__ __launch_bounds__(512) void lstm_kernel(
    const float* __restrict__ qstar, float* __restrict__ h_l,
    float* __restrict__ c_l,
    const float* __restrict__ w_ih, const float* __restrict__ w_hh,
    const float* __restrict__ b_ih, const float* __restrict__ b_hh)
{
    __shared__ float s_gates[4 * HID];
    const int g = blockIdx.x;
    const int j = threadIdx.x;

    float acc = b_ih[j] + b_hh[j];
    const float* q  = qstar + (size_t)g * 2 * HID;
    const float* wi = w_ih + (size_t)j * 2 * HID;
    for (int k = 0; k < 2 * HID; ++k) acc += q[k] * wi[k];
    const float* hl = h_l + (size_t)g * HID;
    const float* wh = w_hh + (size_t)j * HID;
    for (int k = 0; k < HID; ++k) acc += hl[k] * wh[k];
    s_gates[j] = acc;
    __syncthreads();

    if (j < HID) {
        const float i_g = sigmoidf_(s_gates[j]);
        const float f_g = sigmoidf_(s_gates[HID + j]);
        const float g_g = tanhf(s_gates[2 * HID + j]);
        const float o_g = sigmoidf_(s_gates[3 * HID + j]);
        const float c = f_g * c_l[(size_t)g * HID + j] + i_g * g_g;
        c_l[(size_t)g * HID + j] = c;
        h_l[(size_t)g * HID + j] = o_g * tanhf(c);
    }
}

// Attention score e[n] = <h[n], q[batch[n]]>, with fused segment max.
__global__ void attn_e_kernel(
    const float* __restrict__ h, const int* __restrict__ batch,
    const float* __restrict__ h_l, float* __restrict__ e,
    float* __restrict__ emax, int n)
{
    const int i = blockIdx.x * blockDim.x + threadIdx.x;
    if (i >= n) return;
    const int b = batch[i];
    const float4* hv = (const float4*)(h + (size_t)i * HID);
    const float4* qv = (const float4*)(h_l + (size_t)b * HID);
    float s = 0.0f;
#pragma unroll 8
    for (int k = 0; k < HID / 4; ++k) {
        const float4 a = hv[k], q = qv[k];
        s += a.x * q.x + a.y * q.y + a.z * q.z + a.w * q.w;
    }
    e[i] = s;
    atomicMaxF(&emax[b], s);
}

// a[n] = exp(e[n]-emax[b]) ; denom[b] += a[n]
__global__ void attn_a_kernel(
    const float* __restrict__ e, const int* __restrict__ batch,
    const float* __restrict__ emax, float* __restrict__ a,
    float* __restrict__ denom, int n)
{
    const int i = blockIdx.x * blockDim.x + threadIdx.x;
    if (i >= n) return;
    const int b = batch[i];
    const float v = expf(e[i] - emax[b]);
    a[i] = v;
    atomicAdd(&denom[b], v);
}

// num[b,:] += a[n] * h[n,:]  (one wave32 per node, float4 per lane)
__global__ void attn_r_kernel(
    const float* __restrict__ h, const int* __restrict__ batch,
    const float* __restrict__ a, float* __restrict__ num, int n)
{
    const int idx  = blockIdx.x * blockDim.x + threadIdx.x;
    const int node = idx >> 5;
    const int lane = idx & 31;
    if (node >= n) return;
    const int b = batch[node];
    const float av = a[node];
    const float4 v = ((const float4*)(h + (size_t)node * HID))[lane];
    float* o = num + (size_t)b * HID + lane * 4;
    atomicAdd(o + 0, av * v.x);
    atomicAdd(o + 1, av * v.y);
    atomicAdd(o + 2, av * v.z);
    atomicAdd(o + 3, av * v.w);
}

// q_star = [h_l, num/denom]
__global__ void qstar_kernel(
    const float* __restrict__ h_l, const float* __restrict__ num,
    const float* __restrict__ denom, float* __restrict__ qstar)
{
    const int i = blockIdx.x * blockDim.x + threadIdx.x;
    if (i >= NUM_GRAPHS * HID) return;
    const int g = i / HID, d = i % HID;
    const float dn = denom[g];
    const float r = dn > 0.0f ? num[i] / dn : 0.0f;
    qstar[(size_t)g * 2 * HID + d] = h_l[i];
    qstar[(size_t)g * 2 * HID + HID + d] = r;
}

__global__ void pred_kernel(
    const float* __restrict__ qstar, const float* __restrict__ Wp,
    const float* __restrict__ bp, float* __restrict__ out)
{
    const int g = threadIdx.x;
    if (g >= NUM_GRAPHS) return;
    float s = bp[0];
    const float* q = qstar + (size_t)g * 2 * HID;
    for (int k = 0; k < 2 * HID; ++k) s += q[k] * Wp[k];
    out[g] = s;
}

// ---------------------------------------------------------------------------
extern "C" void kernel_launch(void* const* d_in, const int* in_sizes, int n_in,
                              void* d_out, int out_size, void* d_ws, size_t ws_size,
                              hipStream_t stream)
{
    const float* x         = (const float*)d_in[0];
    const int*   edge_idx  = (const int*)d_in[1];
    const int*   batch     = (const int*)d_in[2];
    const float* W_in      = (const float*)d_in[3];
    const float* b_in      = (const float*)d_in[4];
    const float* W_f       = (const float*)d_in[5];
    const float* U_f       = (const float*)d_in[6];
    const float* b_f       = (const float*)d_in[7];
    const float* W_h       = (const float*)d_in[8];
    const float* U_h       = (const float*)d_in[9];
    const float* b_h       = (const float*)d_in[10];
    const float* w_ih      = (const float*)d_in[11];
    const float* w_hh      = (const float*)d_in[12];
    const float* b_ih      = (const float*)d_in[13];
    const float* b_hh      = (const float*)d_in[14];
    const float* W_pred    = (const float*)d_in[15];
    const float* b_pred    = (const float*)d_in[16];

    float* ws    = (float*)d_ws;
    float* h     = ws;                                   // N*HID
    float* m     = h + (size_t)N_NODES * HID;            // N*HID
    float* e     = m + (size_t)N_NODES * HID;            // N
    float* a     = e + N_NODES;                          // N
    float* emax  = a + N_NODES;                          // G
    float* denom = emax + NUM_GRAPHS;                    // G
    float* num   = denom + NUM_GRAPHS;                   // G*HID
    float* qstar = num + NUM_GRAPHS * HID;               // G*2*HID
    float* h_l   = qstar + NUM_GRAPHS * 2 * HID;         // G*HID
    float* c_l   = h_l + NUM_GRAPHS * HID;               // G*HID

    const int NH = N_NODES * HID;
    const float NEG_INF = -__builtin_huge_valf();

    // h = relu(x @ W_in^T + b_in)
    input_kernel<<<N_NODES / 16, 256, 0, stream>>>(x, W_in, b_in, h);

    // MPNN steps
    for (int s = 0; s < MPNN_STEPS; ++s) {
        fill_kernel<<<(NH + 255) / 256, 256, 0, stream>>>(m, 0.0f, NH);
        edge_agg_kernel<<<(N_EDGES * 32 + 255) / 256, 256, 0, stream>>>(
            h, edge_idx, edge_idx + N_EDGES, m, N_EDGES);
        mgu_kernel<<<N_NODES / 16, 256, 0, stream>>>(
            m, h, W_f, U_f, b_f, W_h, U_h, b_h);
    }

    // Set2Set state init
    fill_kernel<<<(NUM_GRAPHS * 2 * HID + 255) / 256, 256, 0, stream>>>(
        qstar, 0.0f, NUM_GRAPHS * 2 * HID);
    fill_kernel<<<(NUM_GRAPHS * HID + 255) / 256, 256, 0, stream>>>(
        h_l, 0.0f, NUM_GRAPHS * HID);
    fill_kernel<<<(NUM_GRAPHS * HID + 255) / 256, 256, 0, stream>>>(
        c_l, 0.0f, NUM_GRAPHS * HID);

    for (int s = 0; s < S2S_STEPS; ++s) {
        lstm_kernel<<<NUM_GRAPHS, 512, 0, stream>>>(
            qstar, h_l, c_l, w_ih, w_hh, b_ih, b_hh);
        fill_kernel<<<1, 256, 0, stream>>>(emax, NEG_INF, NUM_GRAPHS);
        fill_kernel<<<1, 256, 0, stream>>>(denom, 0.0f, NUM_GRAPHS);
        fill_kernel<<<(NUM_GRAPHS * HID + 255) / 256, 256, 0, stream>>>(
            num, 0.0f, NUM_GRAPHS * HID);
        attn_e_kernel<<<(N_NODES + 255) / 256, 256, 0, stream>>>(
            h, batch, h_l, e, emax, N_NODES);
        attn_a_kernel<<<(N_NODES + 255) / 256, 256, 0, stream>>>(
            e, batch, emax, a, denom, N_NODES);
        attn_r_kernel<<<(N_NODES * 32 + 255) / 256, 256, 0, stream>>>(
            h, batch, a, num, N_NODES);
        qstar_kernel<<<(NUM_GRAPHS * HID + 255) / 256, 256, 0, stream>>>(
            h_l, num, denom, qstar);
    }

    pred_kernel<<<1, 128, 0, stream>>>(qstar, W_pred, b_pred, (float*)d_out);
}